// CurveEval2_80779744903773
// MI455X (gfx1250) — compile-verified
//
#include <hip/hip_runtime.h>

namespace {
constexpr int kP          = 3;
constexpr int kM          = 64;          // control points per batch
constexpr int kD          = 3;           // xyz
constexpr int kOut        = 8192;        // samples per batch
constexpr int kKnots      = kM + kP + 1; // 68
constexpr int kPtsPerBlk  = 1024;        // points handled per workgroup
constexpr int kThreads    = 256;         // 8 wave32s
constexpr int kWaves      = kThreads / 32;
constexpr int kGroups     = kPtsPerBlk / 16;           // 64 WMMA groups per block
constexpr int kGrpPerWave = kGroups / kWaves;          // 8
}

typedef __attribute__((ext_vector_type(2))) float v2f;
typedef __attribute__((ext_vector_type(8))) float v8f;

__global__ __launch_bounds__(kThreads)
void bspline_curve_wmma(const float* __restrict__ ctrl_pts,
                        const float* __restrict__ knot_u,
                        float* __restrict__ out)
{
    // LDS staging: knots (68 + pad), zero-padded column-major ctrl [dim(16)][m(64)],
    // and the block's output slab for coalesced writeback.
    __shared__ __align__(16) float kn[kKnots + 12];
    __shared__ __align__(16) float ctrl_cm[16 * kM];
    __shared__ __align__(16) float out_stage[kPtsPerBlk * kD];

    const int tid   = threadIdx.x;
    const int batch = blockIdx.y;
    const int tbase = blockIdx.x * kPtsPerBlk;

    if (tid == 0) {
        __builtin_prefetch(ctrl_pts + (size_t)batch * kM * kD, 0, 0);
        __builtin_prefetch(knot_u   + (size_t)batch * kKnots, 0, 0);
    }

    // ---- stage knots ----
    if (tid < kKnots) kn[tid] = knot_u[(size_t)batch * kKnots + tid];
    else if (tid < kKnots + 12) kn[tid] = 1.0f;   // harmless pad

    // ---- zero the padded ctrl tile, then scatter real values ----
    for (int i = tid; i < 16 * kM; i += kThreads) ctrl_cm[i] = 0.0f;
    __syncthreads();
    if (tid < kM * kD) {
        int m = tid / kD, d = tid % kD;
        ctrl_cm[d * kM + m] = ctrl_pts[((size_t)batch * kM + m) * kD + d];
    }
    __syncthreads();

    const int  lane = tid & 31;
    const int  wave = tid >> 5;
    const int  pm   = lane & 15;        // point within 16-point group == output column
    const bool hiHf = (lane >= 16);
    const int  hi2  = hiHf ? 2 : 0;
    const float step = (1.0f - 2e-5f) / (float)(kOut - 1);

    for (int gi = 0; gi < kGrpPerWave; ++gi) {
        const int group = wave * kGrpPerWave + gi;
        const int g16   = group * 16;
        const int tg    = tbase + g16 + pm;              // global sample index
        const float u   = 1e-5f + step * (float)tg;

        // ---- find span: count knots (from index P) strictly below u - 1e-8 ----
        const float thr = u - 1e-8f;
        int count = 0;
        #pragma unroll 4
        for (int i = 0; i <= kM; ++i)                    // kn[P .. P+64]
            count += (kn[kP + i] < thr) ? 1 : 0;
        const int s = kP + count - 1;                    // uspan, in [3, 63]

        // ---- Cox–de Boor (matches reference fp expression order) ----
        float Nr[kP + 1] = {1.0f, 0.0f, 0.0f, 0.0f};
        #pragma unroll
        for (int k = 1; k <= kP; ++k) {
            float saved = 0.0f;
            #pragma unroll
            for (int r = 0; r < k; ++r) {
                const float U1 = kn[s + r + 1];
                const float U2 = kn[s + 1 - k + r];
                const float dU = (U1 - u) + (u - U2);
                const float temp = (dU == 0.0f) ? 1e-4f : (Nr[r] / dU);
                Nr[r] = saved + (U1 - u) * temp;
                saved = (u - U2) * temp;
            }
            Nr[k] = saved;
        }
        const float N0 = Nr[0], N1 = Nr[1], N2 = Nr[2], N3 = Nr[3];
        const int jlo = s - kP;                          // first nonzero basis column

        // ---- GEMM row-block: D(16x16) += A(16x4) x B(4x16) over K chunks ----
        v8f acc = {};
        for (int c = 0; c < kM / 4; ++c) {
            const int j0 = c * 4;
            // does any lane's support [jlo, s] intersect columns [j0, j0+3]?
            const bool overlap = (s >= j0) && (jlo <= j0 + 3);
            if (__any((int)overlap)) {
                // A fragment (16x4 f32): lanes 0-15 hold {K0,K1}, lanes 16-31 {K2,K3}
                const int ja = j0 + hi2;
                int ra = ja - jlo, rb = ra + 1;
                float a0 = 0.0f, a1 = 0.0f;
                a0 = (ra == 0) ? N0 : a0; a0 = (ra == 1) ? N1 : a0;
                a0 = (ra == 2) ? N2 : a0; a0 = (ra == 3) ? N3 : a0;
                a1 = (rb == 0) ? N0 : a1; a1 = (rb == 1) ? N1 : a1;
                a1 = (rb == 2) ? N2 : a1; a1 = (rb == 3) ? N3 : a1;
                v2f afrag; afrag.x = a0; afrag.y = a1;
                // B fragment (4x16 f32): rows ja, ja+1 at column pm (ja even -> 8B aligned)
                const v2f bfrag = *(const v2f*)&ctrl_cm[pm * kM + ja];
                acc = __builtin_amdgcn_wmma_f32_16x16x4_f32(
                        /*neg_a=*/false, afrag, /*neg_b=*/false, bfrag,
                        /*c_mod=*/(short)0, acc, /*reuse_a=*/false, /*reuse_b=*/false);
            }
        }

        // ---- D layout: VGPR v -> row M = v (+8 for high half), column N = pm ----
        if (pm < kD) {
            const int mb = hiHf ? 8 : 0;
            #pragma unroll
            for (int v = 0; v < 8; ++v)
                out_stage[(g16 + mb + v) * kD + pm] = acc[v];
        }
    }
    __syncthreads();

    // ---- coalesced float4 writeback of the block's 1024x3 slab ----
    {
        const size_t base = ((size_t)batch * kOut + tbase) * kD;   // multiple of 4 floats
        float4*       og = (float4*)(out + base);
        const float4* os = (const float4*)out_stage;
        #pragma unroll
        for (int k = 0; k < (kPtsPerBlk * kD / 4) / kThreads; ++k)
            og[tid + k * kThreads] = os[tid + k * kThreads];
    }
}

extern "C" void kernel_launch(void* const* d_in, const int* in_sizes, int n_in,
                              void* d_out, int out_size, void* d_ws, size_t ws_size,
                              hipStream_t stream) {
    const float* ctrl  = (const float*)d_in[0];   // (B, 64, 3) f32
    const float* knots = (const float*)d_in[1];   // (B, 68)   f32
    float* out = (float*)d_out;                   // (B, 8192, 3) f32
    const int Bn = in_sizes[1] / kKnots;
    dim3 grid(kOut / kPtsPerBlk, Bn);
    bspline_curve_wmma<<<grid, kThreads, 0, stream>>>(ctrl, knots, out);
}